// OntologyNN_75445395521547
// MI455X (gfx1250) — compile-verified
//
#include <hip/hip_runtime.h>

typedef float v2f __attribute__((ext_vector_type(2)));
typedef float v8f __attribute__((ext_vector_type(8)));

#define B_DIM      128
#define N_FEAT     4096
#define N_ANNOT    4096
#define N_NODES    8192
#define NUM_LEVELS 8
#define LVL        512
#define KCH        8
#define EDGES_PER_SAMPLE (NUM_LEVELS * LVL * KCH)   // 32768

// ---------------------------------------------------------------------------
// Kernel 1: h = relu(X @ W^T + b)  via V_WMMA_F32_16X16X4_F32
//   X: [128, 4096]  W: [4096, 4096] (row-major, out x in)  H: [128, 4096]
// One wave computes one 16(batch) x 16(out) tile, looping K in steps of 4.
// A frag (16x4, M x K): lane l<16 -> row M=l, K=0..1 ; lane l>=16 -> K=2..3
// B frag (4x16, K x N): lane l<16 -> col N=l (= weight row), K=0..1 ; etc.
// Both are contiguous float2 loads along the feature axis.
// ---------------------------------------------------------------------------
__global__ __launch_bounds__(256) void fc1_wmma_kernel(
    const float* __restrict__ X,
    const float* __restrict__ W,
    const float* __restrict__ bias,
    float* __restrict__ H)
{
    const int wave   = threadIdx.x >> 5;
    const int lane   = threadIdx.x & 31;
    const int tile   = blockIdx.x * 8 + wave;   // 0..2047
    const int n_tile = tile & 255;              // 256 tiles along out-features
    const int m_tile = tile >> 8;               // 8 tiles along batch
    const int half   = lane >> 4;
    const int lrow   = lane & 15;

    const float* arow = X + (size_t)(m_tile * 16 + lrow) * N_FEAT + 2 * half;
    const float* brow = W + (size_t)(n_tile * 16 + lrow) * N_FEAT + 2 * half;

    v8f acc = {};
    #pragma unroll 8
    for (int k = 0; k < N_FEAT; k += 4) {
        v2f a = *(const v2f*)(arow + k);
        v2f b = *(const v2f*)(brow + k);
        // (neg_a, A, neg_b, B, c_mod, C, reuse_a, reuse_b)
        acc = __builtin_amdgcn_wmma_f32_16x16x4_f32(
                  false, a, false, b, (short)0, acc, false, false);
    }

    const int ocol = n_tile * 16 + lrow;
    const float bv = bias[ocol];
    #pragma unroll
    for (int r = 0; r < 8; ++r) {
        const int m = m_tile * 16 + r + 8 * half;   // C/D layout: VGPR r -> M=r (+8 for hi half)
        float v = acc[r] + bv;
        H[(size_t)m * N_ANNOT + ocol] = v > 0.f ? v : 0.f;
    }
}

// ---------------------------------------------------------------------------
// Kernel 2: layered DAG propagation + fc2 head. One workgroup per sample.
// Node state (1 scalar each, 8192 nodes = 32KB) lives in LDS.
//   out[leaf n]  = h[b,n] * lin_r                       (n < 4096)
//   level k node = tanh(lin_l * sum_{8 children} + lin_l_b)
//   y[b] = sum_n out[n] * fc2_w[n] + fc2_b
// Children indices come from edge_index row 0 (src), laid out
// [b][level][node][child] contiguously; dst is implied by that layout.
// ---------------------------------------------------------------------------
__global__ __launch_bounds__(256) void dag_fc2_kernel(
    const float* __restrict__ H,        // [128, 4096] from kernel 1
    const int*   __restrict__ esrc,     // [B * 32768] src row of edge_index
    const float* __restrict__ lin_l_w,  // [1]
    const float* __restrict__ lin_l_b,  // [1]
    const float* __restrict__ lin_r_w,  // [1]
    const float* __restrict__ fc2_w,    // [8192]
    const float* __restrict__ fc2_b,    // [1]
    float* __restrict__ Y)              // [128]
{
    __shared__ float node[N_NODES];
    __shared__ float red[256];

    const int b = blockIdx.x;
    const int t = threadIdx.x;
    const float lr = lin_r_w[0];
    const float ll = lin_l_w[0];
    const float lb = lin_l_b[0];

    // Leaves: out = h * lin_r  (covers both nz and tanh(0)=0 branches)
    const float* h = H + (size_t)b * N_ANNOT;
    for (int n = t; n < N_ANNOT; n += 256)
        node[n] = h[n] * lr;
    __syncthreads();

    const int  node_off = b * N_NODES;
    const int* es = esrc + (size_t)b * EDGES_PER_SAMPLE;
    for (int k = 0; k < NUM_LEVELS; ++k) {
        for (int i = t; i < LVL; i += 256) {
            const int* e = es + k * (LVL * KCH) + i * KCH;  // 32B coalesced per node
            float s = 0.f;
            #pragma unroll
            for (int j = 0; j < KCH; ++j)
                s += node[e[j] - node_off];
            node[N_ANNOT + k * LVL + i] = tanhf(ll * s + lb);
        }
        __syncthreads();
    }

    // fc2 head: dot(out, fc2_w) + fc2_b
    float partial = 0.f;
    for (int n = t; n < N_NODES; n += 256)
        partial += node[n] * fc2_w[n];
    red[t] = partial;
    __syncthreads();
    #pragma unroll
    for (int s = 128; s > 0; s >>= 1) {
        if (t < s) red[t] += red[t + s];
        __syncthreads();
    }
    if (t == 0) Y[b] = red[0] + fc2_b[0];
}

extern "C" void kernel_launch(void* const* d_in, const int* in_sizes, int n_in,
                              void* d_out, int out_size, void* d_ws, size_t ws_size,
                              hipStream_t stream) {
    const float* X       = (const float*)d_in[0];   // feature_data [128,4096]
    const int*   edge    = (const int*)  d_in[1];   // edge_index   [2, B*32768]
    const float* fc1_w   = (const float*)d_in[2];   // [4096,4096]
    const float* fc1_b   = (const float*)d_in[3];   // [4096]
    const float* lin_l_w = (const float*)d_in[4];   // [1,1]
    const float* lin_l_b = (const float*)d_in[5];   // [1]
    const float* lin_r_w = (const float*)d_in[6];   // [1,1]
    const float* fc2_w   = (const float*)d_in[7];   // [1,8192]
    const float* fc2_b   = (const float*)d_in[8];   // [1]

    float* H = (float*)d_ws;                        // [128,4096] = 2 MB scratch

    // 2048 tiles of 16x16, 8 waves (256 threads) per block -> 256 blocks
    fc1_wmma_kernel<<<256, 256, 0, stream>>>(X, fc1_w, fc1_b, H);

    // One workgroup per sample; esrc = row 0 of edge_index
    dag_fc2_kernel<<<B_DIM, 256, 0, stream>>>(H, edge, lin_l_w, lin_l_b,
                                              lin_r_w, fc2_w, fc2_b,
                                              (float*)d_out);
}